// SemanticCodebookLoss_59785944760544
// MI455X (gfx1250) — compile-verified
//
#include <hip/hip_runtime.h>
#include <hip/hip_bf16.h>
#include <stdint.h>

// ---------------------------------------------------------------------------
// Problem constants (from reference): B=16, T=1024, K=1024, D=256
// ---------------------------------------------------------------------------
#define BT      16384      // B*T rows of logits
#define KDIM    1024       // codebook entries / logits last dim
#define DDIM    256        // codebook vector dim
#define HALF_N  8388608u   // (BT*KDIM)/2 : JAX threefry counter pairing offset
#define PAIRS   8192       // BT/2 row pairs

typedef float v2f __attribute__((ext_vector_type(2)));
typedef float v8f __attribute__((ext_vector_type(8)));

// ---------------------------------------------------------------------------
// Threefry-2x32-20 with key = jax.random.key(42) -> (k0,k1) = (0,42)
// Rotation constants per the Threefry spec (matches JAX's implementation).
// ---------------------------------------------------------------------------
__device__ __forceinline__ uint32_t rotl32(uint32_t x, int r) {
    return (x << r) | (x >> (32 - r));   // lowers to v_alignbit_b32
}

__device__ __forceinline__ void threefry2x32_20(uint32_t c0, uint32_t c1,
                                                uint32_t& o0, uint32_t& o1) {
    const uint32_t k0 = 0u, k1 = 42u;
    const uint32_t k2 = k0 ^ k1 ^ 0x1BD11BDAu;
    uint32_t x0 = c0 + k0;
    uint32_t x1 = c1 + k1;
#define TF_R(r) { x0 += x1; x1 = rotl32(x1, r); x1 ^= x0; }
    TF_R(13) TF_R(15) TF_R(26) TF_R(6)   x0 += k1; x1 += k2 + 1u;
    TF_R(17) TF_R(29) TF_R(16) TF_R(24)  x0 += k2; x1 += k0 + 2u;
    TF_R(13) TF_R(15) TF_R(26) TF_R(6)   x0 += k0; x1 += k1 + 3u;
    TF_R(17) TF_R(29) TF_R(16) TF_R(24)  x0 += k1; x1 += k2 + 4u;
    TF_R(13) TF_R(15) TF_R(26) TF_R(6)   x0 += k2; x1 += k0 + 5u;
#undef TF_R
    o0 = x0; o1 = x1;
}

// bits -> uniform [0,1) exactly like jax.random.uniform, then Gumbel(0,1)
__device__ __forceinline__ float bits_to_gumbel(uint32_t bits) {
    float u = __uint_as_float((bits >> 9) | 0x3f800000u) - 1.0f; // [0,1)
    u = fmaf(u, 1.0f - 1e-20f, 1e-20f);                          // minval..maxval
    u = fmaxf(u, 1e-20f);
    return -__logf(-__logf(u));
}

// ---------------------------------------------------------------------------
// Kernel 1: fused Gumbel-noise + argmax over K.
// One wave (32 lanes) owns the row PAIR (p, p+8192): a single threefry call
// per k produces the noise for BOTH rows (JAX pairs counter i with i+N/2),
// halving the integer-ALU cost, which is the phase's compute bottleneck
// (memory side is a 64MB stream -> ~2.7us at 23.3 TB/s).
// Grid: 1024 blocks x 256 threads (8 waves/block).
// ---------------------------------------------------------------------------
__global__ void gumbel_argmax_kernel(const float* __restrict__ logits,
                                     int* __restrict__ predIdx) {
    const int lane = threadIdx.x & 31;
    const int wave = threadIdx.x >> 5;
    const int p    = blockIdx.x * 8 + wave;          // 0..8191
    const float* rowA = logits + (size_t)p * KDIM;
    const float* rowB = logits + (size_t)(p + PAIRS) * KDIM;
    const uint32_t baseA = (uint32_t)p * (uint32_t)KDIM;

    float bestA = -INFINITY, bestB = -INFINITY;
    int   idxA  = 0,         idxB  = 0;

    for (int it = 0; it < 8; ++it) {
        const int c = it * 128 + lane * 4;
        float4 la = *(const float4*)(rowA + c);
        float4 lb = *(const float4*)(rowB + c);
        // speculative prefetch of the next stream chunk (global_prefetch_b8)
        __builtin_prefetch(rowA + c + 512, 0, 0);
        __builtin_prefetch(rowB + c + 512, 0, 0);
        const float va[4] = {la.x, la.y, la.z, la.w};
        const float vb[4] = {lb.x, lb.y, lb.z, lb.w};
#pragma unroll
        for (int j = 0; j < 4; ++j) {
            const int k = c + j;
            uint32_t e = baseA + (uint32_t)k;
            uint32_t x0, x1;
            threefry2x32_20(e, e + HALF_N, x0, x1);
            const float sa = va[j] + bits_to_gumbel(x0);
            const float sb = vb[j] + bits_to_gumbel(x1);
            if (sa > bestA || (sa == bestA && k < idxA)) { bestA = sa; idxA = k; }
            if (sb > bestB || (sb == bestB && k < idxB)) { bestB = sb; idxB = k; }
        }
    }
    // wave32 argmax reduction (first-occurrence tie-break like jnp.argmax)
#pragma unroll
    for (int off = 16; off; off >>= 1) {
        float fa = __shfl_xor(bestA, off, 32); int ia = __shfl_xor(idxA, off, 32);
        if (fa > bestA || (fa == bestA && ia < idxA)) { bestA = fa; idxA = ia; }
        float fb = __shfl_xor(bestB, off, 32); int ib = __shfl_xor(idxB, off, 32);
        if (fb > bestB || (fb == bestB && ib < idxB)) { bestB = fb; idxB = ib; }
    }
    if (lane == 0) {
        predIdx[p]         = idxA;
        predIdx[p + PAIRS] = idxB;
    }
}

// ---------------------------------------------------------------------------
// Kernel 2: Gram = C * C^T (1024x1024, K=256) via V_WMMA_F32_16X16X4_F32.
// One wave per 16x16 output tile; 64 WMMAs per tile (K stepped by 4).
// A 16x4 layout (ISA 7.12.2): lane m: v0=K0, v1=K1 ; lane m+16: v0=K2, v1=K3.
// C/D layout: VGPR r, lanes 0-15 -> (M=r, N=lane); lanes 16-31 -> (M=r+8).
// Grid: 512 blocks x 256 threads (8 waves = 8 tiles per block).
// ---------------------------------------------------------------------------
__global__ void gram_wmma_kernel(const float* __restrict__ C,
                                 float* __restrict__ G) {
    const int lane = threadIdx.x & 31;
    const int wave = threadIdx.x >> 5;
    const int t    = blockIdx.x * 8 + wave;      // 0..4095
    const int i0   = (t >> 6) * 16;
    const int j0   = (t & 63) * 16;
    const int ln   = lane & 15;
    const bool hi  = lane >= 16;

    const float* rowa = C + (size_t)(i0 + ln) * DDIM;
    const float* rowb = C + (size_t)(j0 + ln) * DDIM;

    v8f acc = {0.f, 0.f, 0.f, 0.f, 0.f, 0.f, 0.f, 0.f};
    for (int k = 0; k < DDIM; k += 4) {
        float4 ca = *(const float4*)(rowa + k);
        float4 cb = *(const float4*)(rowb + k);
        v2f a, b;
        a.x = hi ? ca.z : ca.x;  a.y = hi ? ca.w : ca.y;
        b.x = hi ? cb.z : cb.x;  b.y = hi ? cb.w : cb.y;
        // 8 args: (neg_a, A, neg_b, B, c_mod, C, reuse_a, reuse_b)
        acc = __builtin_amdgcn_wmma_f32_16x16x4_f32(
                  false, a, false, b, (short)0, acc, false, false);
    }
    float* gout = G + (size_t)(i0 + (hi ? 8 : 0)) * KDIM + (j0 + ln);
#pragma unroll
    for (int r = 0; r < 8; ++r)
        gout[(size_t)r * KDIM] = acc[r];
}

// ---------------------------------------------------------------------------
// Kernel 3a (Gram path): per position read only G[p,p], G[q,q], G[p,q]
// (12 bytes instead of a 2KB gather). Deterministic block-tree reduction.
// Grid: 64 blocks x 256 threads.
// ---------------------------------------------------------------------------
__global__ void gram_lookup_loss_kernel(const float* __restrict__ G,
                                        const int* __restrict__ pred,
                                        const int* __restrict__ tgt,
                                        float* __restrict__ partials) {
    __shared__ float smem[8];
    const int i = blockIdx.x * 256 + threadIdx.x;   // 0..16383
    const int p = pred[i];
    const int q = tgt[i];
    float v = G[(size_t)p * KDIM + p] + G[(size_t)q * KDIM + q]
              - 2.0f * G[(size_t)p * KDIM + q];
#pragma unroll
    for (int off = 16; off; off >>= 1) v += __shfl_xor(v, off, 32);
    if ((threadIdx.x & 31) == 0) smem[threadIdx.x >> 5] = v;
    __syncthreads();
    if (threadIdx.x == 0) {
        float s = 0.f;
#pragma unroll
        for (int w = 0; w < 8; ++w) s += smem[w];
        partials[blockIdx.x] = s;
    }
}

// ---------------------------------------------------------------------------
// Kernel 3b (fallback if workspace too small for the 4MB Gram):
// direct gather of both codebook rows + squared diff. One wave per position.
// Grid: 2048 blocks x 256 threads.
// ---------------------------------------------------------------------------
__global__ void gather_diff_loss_kernel(const float* __restrict__ C,
                                        const int* __restrict__ pred,
                                        const int* __restrict__ tgt,
                                        float* __restrict__ partials) {
    __shared__ float smem[8];
    const int lane = threadIdx.x & 31;
    const int wave = threadIdx.x >> 5;
    const int i    = blockIdx.x * 8 + wave;
    const float4* cp = (const float4*)(C + (size_t)pred[i] * DDIM);
    const float4* cq = (const float4*)(C + (size_t)tgt[i]  * DDIM);
    float s = 0.f;
#pragma unroll
    for (int j = lane; j < DDIM / 4; j += 32) {
        float4 a = cp[j], b = cq[j];
        float dx = a.x - b.x, dy = a.y - b.y, dz = a.z - b.z, dw = a.w - b.w;
        s += dx * dx + dy * dy + dz * dz + dw * dw;
    }
#pragma unroll
    for (int off = 16; off; off >>= 1) s += __shfl_xor(s, off, 32);
    if (lane == 0) smem[wave] = s;
    __syncthreads();
    if (threadIdx.x == 0) {
        float t = 0.f;
#pragma unroll
        for (int w = 0; w < 8; ++w) t += smem[w];
        partials[blockIdx.x] = t;
    }
}

// ---------------------------------------------------------------------------
// Kernel 4: final deterministic reduction of partials, write mean.
// Single block of 256 threads.
// ---------------------------------------------------------------------------
__global__ void finalize_kernel(const float* __restrict__ partials, int n,
                                float scale, float* __restrict__ out) {
    __shared__ float smem[8];
    float s = 0.f;
    for (int i = threadIdx.x; i < n; i += 256) s += partials[i];
#pragma unroll
    for (int off = 16; off; off >>= 1) s += __shfl_xor(s, off, 32);
    if ((threadIdx.x & 31) == 0) smem[threadIdx.x >> 5] = s;
    __syncthreads();
    if (threadIdx.x == 0) {
        float t = 0.f;
#pragma unroll
        for (int w = 0; w < 8; ++w) t += smem[w];
        out[0] = t * scale;
    }
}

// ---------------------------------------------------------------------------
// Launch
// ---------------------------------------------------------------------------
extern "C" void kernel_launch(void* const* d_in, const int* in_sizes, int n_in,
                              void* d_out, int out_size, void* d_ws, size_t ws_size,
                              hipStream_t stream) {
    const float* logits   = (const float*)d_in[0];   // [16,1024,1024] f32
    const float* codebook = (const float*)d_in[1];   // [1024,256]    f32
    const int*   tgt      = (const int*)d_in[2];     // [16,1024]     i32
    float*       out      = (float*)d_out;           // scalar f32

    // workspace layout
    int*   predIdx  = (int*)d_ws;                                 // 64 KB
    float* partials = (float*)((char*)d_ws + 65536);              // 8 KB
    float* gram     = (float*)((char*)d_ws + 65536 + 8192);       // 4 MB
    const size_t need_gram = 65536 + 8192 + (size_t)KDIM * KDIM * 4;

    const float scale = 1.0f / ((float)BT * (float)DDIM);

    // Phase 1: gumbel noise + argmax (streams the 64MB logits once)
    gumbel_argmax_kernel<<<PAIRS / 8, 256, 0, stream>>>(logits, predIdx);

    if (ws_size >= need_gram) {
        // Phase 2: Gram matrix on the WMMA pipe, then 12B/position lookups
        gram_wmma_kernel<<<512, 256, 0, stream>>>(codebook, gram);
        gram_lookup_loss_kernel<<<BT / 256, 256, 0, stream>>>(gram, predIdx, tgt, partials);
        finalize_kernel<<<1, 256, 0, stream>>>(partials, BT / 256, scale, out);
    } else {
        // Fallback: direct gather + squared-diff
        gather_diff_loss_kernel<<<BT / 8, 256, 0, stream>>>(codebook, predIdx, tgt, partials);
        finalize_kernel<<<1, 256, 0, stream>>>(partials, BT / 8, scale, out);
    }
}